// Transform_23261542875637
// MI455X (gfx1250) — compile-verified
//
#include <hip/hip_runtime.h>
#include <math.h>

// Problem constants from setup_inputs():
//   x: [8, 32, 32, 64, 64] f32, d_out=48, h_out=96, w_out=96, variance=1
#define NDIM   8
#define CDIM   32
#define DIN    32
#define HIN    64
#define WIN    64
#define DOUT   48
#define HOUT   96
#define WOUT   96
#define TD     4            // output d-slices per workgroup (48/4 = 12 tiles)
#define NTHREADS 256        // 8 waves of 32 on CDNA5
#define PLANE  (HIN * WIN)  // 4096 floats = 16 KB
#define MAXPL  4            // worst-case staged z-planes (span of 4 d_out <= 4)

typedef int v4i __attribute__((vector_size(16)));
typedef v4i __attribute__((address_space(1)))* gv4i_ptr;  // global (AS1)
typedef v4i __attribute__((address_space(3)))* lv4i_ptr;  // LDS    (AS3)

__device__ __forceinline__ void wait_async0() {
#if __has_builtin(__builtin_amdgcn_s_wait_asynccnt)
  __builtin_amdgcn_s_wait_asynccnt(0);
#else
  asm volatile("s_wait_asynccnt 0" ::: "memory");
#endif
}

__global__ __launch_bounds__(NTHREADS)
void resize3d_var_kernel(const float* __restrict__ x, float* __restrict__ out) {
  __shared__ float tile[MAXPL * PLANE];   // 64 KB of the 320 KB WGP LDS

  const int tid   = threadIdx.x;
  const int wg    = blockIdx.x;
  const int dtile = wg % (DOUT / TD);
  const int nc    = wg / (DOUT / TD);     // n*CDIM + c
  const int d0    = dtile * TD;

  const float SZ = (float)(DIN - 1) / (float)(DOUT - 1);   // 31/47
  const float SY = (float)(HIN - 1) / (float)(HOUT - 1);   // 63/95
  const float SX = (float)(WIN - 1) / (float)(WOUT - 1);   // 63/95

  // z-plane range needed by this d-tile (same float math as the inner loop).
  int z_lo = (int)floorf((float)d0 * SZ);
  int z_hi = (int)floorf((float)(d0 + TD - 1) * SZ) + 1;
  if (z_hi > DIN - 1)          z_hi = DIN - 1;
  if (z_hi > z_lo + MAXPL - 1) z_hi = z_lo + MAXPL - 1;   // safety (never hit)
  const int npl  = z_hi - z_lo + 1;
  const int nvec = npl * (PLANE / 4);                      // float4 count

  const float* src = x + ((size_t)nc * DIN + z_lo) * PLANE;

#if __has_builtin(__builtin_amdgcn_global_load_async_to_lds_b128)
  // --- CDNA5 async staging: global -> LDS without touching VGPR data ---
  {
    gv4i_ptr g = (gv4i_ptr)src;    // builtin wants non-const AS1 v4i*
    lv4i_ptr l = (lv4i_ptr)tile;   // AS3 v4i*
    for (int i = tid; i < nvec; i += NTHREADS) {
      __builtin_amdgcn_global_load_async_to_lds_b128(g + i, l + i, 0, 0);
    }
    wait_async0();       // each wave waits for its own ASYNCcnt,
    __syncthreads();     // then barrier -> all planes resident in LDS

    // In-place transform: apply exp(x/2) once per staged element.
    float4* t4 = (float4*)tile;
    for (int i = tid; i < nvec; i += NTHREADS) {
      float4 v = t4[i];
      v.x = __expf(v.x * 0.5f);
      v.y = __expf(v.y * 0.5f);
      v.z = __expf(v.z * 0.5f);
      v.w = __expf(v.w * 0.5f);
      t4[i] = v;
    }
  }
#else
  // Fallback: direct staged copy with exp applied on the way in.
  {
    const float4* s4 = (const float4*)src;
    float4*       t4 = (float4*)tile;
    for (int i = tid; i < nvec; i += NTHREADS) {
      float4 v = s4[i];
      v.x = __expf(v.x * 0.5f);
      v.y = __expf(v.y * 0.5f);
      v.z = __expf(v.z * 0.5f);
      v.w = __expf(v.w * 0.5f);
      t4[i] = v;
    }
  }
#endif
  __syncthreads();

  // --- produce TD x 96 x 96 outputs from LDS ---
  const size_t obase = ((size_t)nc * DOUT + d0) * (size_t)(HOUT * WOUT);
  for (int dl = 0; dl < TD; ++dl) {
    const int   d_o = d0 + dl;
    const float fz  = (float)d_o * SZ;
    const int   z0  = (int)floorf(fz);
    const float tz  = fz - (float)z0;
    const int   z1  = (z0 + 1 > DIN - 1) ? (DIN - 1) : (z0 + 1);
    const int   zb0 = (z0 - z_lo) * PLANE;
    const int   zb1 = (z1 - z_lo) * PLANE;

    for (int j = tid; j < HOUT * WOUT; j += NTHREADS) {   // 36 iters exactly
      const int   h_o = j / WOUT;
      const int   w_o = j - h_o * WOUT;

      const float fy = (float)h_o * SY;
      const int   y0 = (int)fy;                 // fy >= 0 -> trunc == floor
      const float ty = fy - (float)y0;
      const int   y1 = (y0 + 1 > HIN - 1) ? (HIN - 1) : (y0 + 1);

      const float fx = (float)w_o * SX;
      const int   x0 = (int)fx;
      const float tx = fx - (float)x0;
      const int   x1 = (x0 + 1 > WIN - 1) ? (WIN - 1) : (x0 + 1);

      const int r0 = y0 * WIN, r1 = y1 * WIN;
      const float v000 = tile[zb0 + r0 + x0];
      const float v001 = tile[zb0 + r0 + x1];
      const float v010 = tile[zb0 + r1 + x0];
      const float v011 = tile[zb0 + r1 + x1];
      const float v100 = tile[zb1 + r0 + x0];
      const float v101 = tile[zb1 + r0 + x1];
      const float v110 = tile[zb1 + r1 + x0];
      const float v111 = tile[zb1 + r1 + x1];

      const float a0 = v000 + tx * (v001 - v000);
      const float a1 = v010 + tx * (v011 - v010);
      const float b0 = v100 + tx * (v101 - v100);
      const float b1 = v110 + tx * (v111 - v110);
      const float c0 = a0 + ty * (a1 - a0);
      const float c1 = b0 + ty * (b1 - b0);
      const float v  = c0 + tz * (c1 - c0);

      const float r = 2.0f * __logf(v);         // variance=1 path
      // NT store: 453 MB output stream must not evict the L2-resident input.
      __builtin_nontemporal_store(r, &out[obase + (size_t)dl * (HOUT * WOUT) + j]);
    }
  }
}

extern "C" void kernel_launch(void* const* d_in, const int* in_sizes, int n_in,
                              void* d_out, int out_size, void* d_ws, size_t ws_size,
                              hipStream_t stream) {
  (void)in_sizes; (void)n_in; (void)d_ws; (void)ws_size; (void)out_size;
  const float* x   = (const float*)d_in[0];
  float*       out = (float*)d_out;
  const int nblocks = NDIM * CDIM * (DOUT / TD);   // 3072
  resize3d_var_kernel<<<nblocks, NTHREADS, 0, stream>>>(x, out);
}